// RNNCell_61718680043841
// MI455X (gfx1250) — compile-verified
//
#include <hip/hip_runtime.h>
#include <hip/hip_bf16.h>

typedef __attribute__((ext_vector_type(2))) float v2f;
typedef __attribute__((ext_vector_type(8))) float v8f;

// ---------------------------------------------------------------------------
// fp32 WMMA GEMM:  C(M,N) = A(M,K) @ W(N,K)^T [+ bias(N)]
// Block tile 64(M) x 128(N), 256 threads = 8 waves (4 along M, 2 along N).
// Each wave computes a 16x64 strip as four 16x16 V_WMMA_F32_16X16X4_F32 accums.
// Requires: M % 64 == 0, N % 128 == 0, K % 16 == 0 (true for all call sites).
// ---------------------------------------------------------------------------
#define TM 64
#define TN 128
#define TK 16
#define LDP (TK + 4)   // LDS row pitch (floats): keeps 16B alignment, spreads banks

__global__ __launch_bounds__(256) void gemm_wmma_f32(
    const float* __restrict__ A, int lda,
    const float* __restrict__ W, int ldw,
    const float* __restrict__ bias,
    float* __restrict__ C, int ldc,
    int K)
{
    __shared__ float As[TM * LDP];
    __shared__ float Ws[TN * LDP];

    const int tid  = threadIdx.x;
    const int wave = tid >> 5;
    const int lane = tid & 31;
    const int m    = lane & 15;   // column-of-tile for B/C, row-of-tile for A
    const int half = lane >> 4;

    const int m_off = (wave & 3) * 16;   // wave's M offset within block tile
    const int n_off = (wave >> 2) * 64;  // wave's N offset within block tile

    const int row0 = blockIdx.y * TM;
    const int col0 = blockIdx.x * TN;

    v8f acc[4] = {};

    // cooperative loaders
    const int arow = tid >> 2;          // 0..63
    const int akq  = (tid & 3) * 4;     // 0,4,8,12
    const int wrow = tid >> 1;          // 0..127
    const int wkq  = (tid & 1) * 8;     // 0,8

    for (int k0 = 0; k0 < K; k0 += TK) {
        // A block: 64 x 16
        float4 a4 = *(const float4*)(A + (size_t)(row0 + arow) * lda + k0 + akq);
        *(float4*)(&As[arow * LDP + akq]) = a4;
        // W block: 128 x 16 (W is N x K row-major; we need W[n][k])
        const float* wp = W + (size_t)(col0 + wrow) * ldw + k0 + wkq;
        float4 w0 = *(const float4*)(wp);
        float4 w1 = *(const float4*)(wp + 4);
        *(float4*)(&Ws[wrow * LDP + wkq])     = w0;
        *(float4*)(&Ws[wrow * LDP + wkq + 4]) = w1;
        __syncthreads();

        #pragma unroll
        for (int kk = 0; kk < TK; kk += 4) {
            const int ke = kk + 2 * half;  // this lane's k-pair (ISA f32 A/B layout)
            float2 af = *(const float2*)(&As[(m_off + m) * LDP + ke]);
            v2f a; a.x = af.x; a.y = af.y;
            #pragma unroll
            for (int j = 0; j < 4; ++j) {
                float2 bf = *(const float2*)(&Ws[(n_off + 16 * j + m) * LDP + ke]);
                v2f b; b.x = bf.x; b.y = bf.y;
                acc[j] = __builtin_amdgcn_wmma_f32_16x16x4_f32(
                    false, a, false, b, (short)0, acc[j], false, false);
            }
        }
        __syncthreads();
    }

    // epilogue: C layout — VGPR i holds row (8*half + i), col = m
    #pragma unroll
    for (int j = 0; j < 4; ++j) {
        const int c  = col0 + n_off + 16 * j + m;
        const float bb = bias ? bias[c] : 0.0f;
        #pragma unroll
        for (int i = 0; i < 8; ++i) {
            const int r = row0 + m_off + 8 * half + i;
            C[(size_t)r * ldc + c] = acc[j][i] + bb;
        }
    }
}

// ---------------------------------------------------------------------------
// Prep: spk_id = argmax(speaker_mask), sel0 gather, build
// fspk = [feature | sel0], and the feature half of fctx.
// One block per b, 256 threads.
// ---------------------------------------------------------------------------
__global__ __launch_bounds__(256) void prep_kernel(
    const float* __restrict__ feature,   // (B,F)
    const float* __restrict__ mask,      // (B,P)
    const float* __restrict__ speaker0,  // (B,P,S)
    int* __restrict__ spk_id,            // (B)
    float* __restrict__ sel0,            // (B,S)
    float* __restrict__ fspk,            // (B,F+S)
    float* __restrict__ fctx,            // (B,F+G), G==S
    int F_, int S_, int P_)
{
    const int b = blockIdx.x;
    const int tid = threadIdx.x;

    int best = 0; float bv = mask[(size_t)b * P_];
    for (int p = 1; p < P_; ++p) {
        float mv = mask[(size_t)b * P_ + p];
        if (mv > bv) { bv = mv; best = p; }
    }
    if (tid == 0) spk_id[b] = best;

    const int FS = F_ + S_;
    for (int i = tid; i < F_; i += 256) {
        float fv = feature[(size_t)b * F_ + i];
        fspk[(size_t)b * FS + i] = fv;
        fctx[(size_t)b * FS + i] = fv;
    }
    const float* src = speaker0 + ((size_t)b * P_ + best) * S_;
    for (int i = tid; i < S_; i += 256) {
        float sv = src[i];
        sel0[(size_t)b * S_ + i] = sv;
        fspk[(size_t)b * FS + F_ + i] = sv;
    }
}

// ---------------------------------------------------------------------------
// Fused attention: scores -> softmax -> context, one block per b.
// Writes context into the tail of fctx[b].
// ---------------------------------------------------------------------------
__global__ __launch_bounds__(256) void attn_kernel(
    const float* __restrict__ xw,   // (B,G) = feature @ attn_W^T
    const float* __restrict__ GH,   // (T,B,G)
    float* __restrict__ fctx,       // (B,F+G): write cols [F, F+G)
    int B_, int T_, int G_, int F_)
{
    const int b = blockIdx.x;
    const int tid = threadIdx.x;
    __shared__ float xs[512];
    __shared__ float sc[256];
    __shared__ float red[256];

    xs[tid]       = xw[(size_t)b * G_ + tid];
    xs[tid + 256] = xw[(size_t)b * G_ + tid + 256];
    __syncthreads();

    const int wave = tid >> 5, lane = tid & 31;
    for (int t = wave; t < T_; t += 8) {
        const float* gh = GH + ((size_t)t * B_ + b) * G_;
        float s = 0.f;
        for (int g = lane; g < G_; g += 32) s += xs[g] * gh[g];
        for (int off = 16; off > 0; off >>= 1) s += __shfl_xor(s, off, 32);
        if (lane == 0) sc[t] = s;
    }
    __syncthreads();

    // softmax over T=256
    float v = sc[tid];
    red[tid] = v;
    __syncthreads();
    for (int off = 128; off > 0; off >>= 1) {
        if (tid < off) red[tid] = fmaxf(red[tid], red[tid + off]);
        __syncthreads();
    }
    const float mx = red[0];
    __syncthreads();
    float e = __expf(v - mx);
    red[tid] = e;
    __syncthreads();
    for (int off = 128; off > 0; off >>= 1) {
        if (tid < off) red[tid] += red[tid + off];
        __syncthreads();
    }
    const float inv = 1.0f / red[0];
    __syncthreads();
    sc[tid] = e * inv;   // alpha[b][tid]
    __syncthreads();

    // context: thread owns g = 2*tid, 2*tid+1
    float cx = 0.f, cy = 0.f;
    for (int t = 0; t < T_; ++t) {
        const float a = sc[t];
        const float2 gv = *(const float2*)(GH + ((size_t)t * B_ + b) * G_ + 2 * tid);
        cx += a * gv.x; cy += a * gv.y;
    }
    float2 c2; c2.x = cx; c2.y = cy;
    *(float2*)(fctx + (size_t)b * (F_ + G_) + F_ + 2 * tid) = c2;
}

// ---------------------------------------------------------------------------
// GRU gate fusion: out = (1-z)*n + z*h   (gi/gh are (M,3N), biases pre-added)
// ---------------------------------------------------------------------------
__global__ __launch_bounds__(256) void gru_gate_kernel(
    const float* __restrict__ gi, const float* __restrict__ gh,
    const float* __restrict__ h, float* __restrict__ out, int N_)
{
    const int idx = blockIdx.x * 256 + threadIdx.x;
    const int r = idx / N_, c = idx % N_;
    const size_t gb = (size_t)r * 3 * N_;
    float ir = gi[gb + c], iz = gi[gb + N_ + c], in = gi[gb + 2 * N_ + c];
    float hr = gh[gb + c], hz = gh[gb + N_ + c], hn = gh[gb + 2 * N_ + c];
    float rr = 1.f / (1.f + __expf(-(ir + hr)));
    float zz = 1.f / (1.f + __expf(-(iz + hz)));
    float nn = tanhf(in + rr * hn);
    float hv = h[(size_t)r * N_ + c];
    out[(size_t)r * N_ + c] = (1.f - zz) * nn + zz * hv;
}

// ---------------------------------------------------------------------------
// speaker = speaker0 with row (b, spk_id[b]) replaced by parties_sel[b].
// One block per (b,p).
// ---------------------------------------------------------------------------
__global__ __launch_bounds__(256) void speaker_out_kernel(
    const float* __restrict__ speaker0, const float* __restrict__ parties,
    const int* __restrict__ spk_id, float* __restrict__ out, int P_, int S_)
{
    const int bp = blockIdx.x;
    const int b = bp / P_, p = bp % P_;
    const int tid = threadIdx.x;
    const float* src = (spk_id[b] == p) ? (parties + (size_t)b * S_)
                                        : (speaker0 + (size_t)bp * S_);
    float* dst = out + (size_t)bp * S_;
    for (int i = tid; i < S_; i += 256) dst[i] = src[i];
}

// ---------------------------------------------------------------------------
extern "C" void kernel_launch(void* const* d_in, const int* in_sizes, int n_in,
                              void* d_out, int out_size, void* d_ws, size_t ws_size,
                              hipStream_t stream)
{
    const float* feature  = (const float*)d_in[0];
    const float* mask     = (const float*)d_in[1];
    const float* GH       = (const float*)d_in[2];
    const float* speaker0 = (const float*)d_in[3];
    const float* emotion0 = (const float*)d_in[4];
    const float* Wg_ih = (const float*)d_in[5];
    const float* Wg_hh = (const float*)d_in[6];
    const float* bg_ih = (const float*)d_in[7];
    const float* bg_hh = (const float*)d_in[8];
    const float* Wp_ih = (const float*)d_in[9];
    const float* Wp_hh = (const float*)d_in[10];
    const float* bp_ih = (const float*)d_in[11];
    const float* bp_hh = (const float*)d_in[12];
    const float* We_ih = (const float*)d_in[13];
    const float* We_hh = (const float*)d_in[14];
    const float* be_ih = (const float*)d_in[15];
    const float* be_hh = (const float*)d_in[16];
    const float* attnW = (const float*)d_in[17];

    constexpr int F = 1024, G = 512, S = 512, E = 256, P = 8, B = 256, T = 256;

    float* out         = (float*)d_out;
    float* emotion_out = out;                    // (B,E)
    float* gstate_out  = out + (size_t)B * E;    // (B,G)
    float* speaker_out = gstate_out + (size_t)B * G; // (B,P,S)

    float* ws    = (float*)d_ws;
    int*   spk   = (int*)ws;                     // 256 ints
    float* sel0  = ws + 256;                     // (B,S)
    float* fspk  = sel0 + (size_t)B * S;         // (B,F+S)
    float* fctx  = fspk + (size_t)B * (F + S);   // (B,F+G)
    float* xw    = fctx + (size_t)B * (F + G);   // (B,G)
    float* pst   = xw   + (size_t)B * G;         // (B,S) parties_sel
    float* gi    = pst  + (size_t)B * S;         // (B,3*G) max
    float* gh    = gi   + (size_t)B * 3 * G;     // (B,3*G) max

    const dim3 blk(256);
    const float* hlast = GH + (size_t)(T - 1) * B * G;

    // 1) prep: spk_id, sel0, fspk=[feature|sel0], feature half of fctx
    prep_kernel<<<B, blk, 0, stream>>>(feature, mask, speaker0, spk, sel0, fspk, fctx, F, S, P);

    // 2) global GRU
    const dim3 gN1536(1536 / TN, B / TM);
    gemm_wmma_f32<<<gN1536, blk, 0, stream>>>(fspk, F + S, Wg_ih, F + S, bg_ih, gi, 3 * G, F + S);
    gemm_wmma_f32<<<gN1536, blk, 0, stream>>>(hlast, G, Wg_hh, G, bg_hh, gh, 3 * G, G);
    gru_gate_kernel<<<(B * G) / 256, blk, 0, stream>>>(gi, gh, hlast, gstate_out, G);

    // 3) attention: x_ = feature @ attn_W^T, then fused scores/softmax/context
    const dim3 gN512(512 / TN, B / TM);
    gemm_wmma_f32<<<gN512, blk, 0, stream>>>(feature, F, attnW, F, nullptr, xw, G, F);
    attn_kernel<<<B, blk, 0, stream>>>(xw, GH, fctx, B, T, G, F);

    // 4) parties GRU — only the spk_id-selected rows matter (h = sel0)
    gemm_wmma_f32<<<gN1536, blk, 0, stream>>>(fctx, F + G, Wp_ih, F + G, bp_ih, gi, 3 * S, F + G);
    gemm_wmma_f32<<<gN1536, blk, 0, stream>>>(sel0, S, Wp_hh, S, bp_hh, gh, 3 * S, S);
    gru_gate_kernel<<<(B * S) / 256, blk, 0, stream>>>(gi, gh, sel0, pst, S);

    // 5) speaker output: copy speaker0, replacing selected rows with parties_sel
    speaker_out_kernel<<<B * P, blk, 0, stream>>>(speaker0, pst, spk, speaker_out, P, S);

    // 6) emotion GRU
    const dim3 gN768(768 / TN, B / TM);
    gemm_wmma_f32<<<gN768, blk, 0, stream>>>(pst, S, We_ih, S, be_ih, gi, 3 * E, S);
    gemm_wmma_f32<<<gN768, blk, 0, stream>>>(emotion0, E, We_hh, E, be_hh, gh, 3 * E, E);
    gru_gate_kernel<<<(B * E) / 256, blk, 0, stream>>>(gi, gh, emotion0, emotion_out, E);

    (void)in_sizes; (void)n_in; (void)out_size; (void)ws_size;
}